// HOGLayerC_56642028700214
// MI455X (gfx1250) — compile-verified
//
#include <hip/hip_runtime.h>

// ---------------- problem constants (fixed by setup_inputs) ----------------
#define BATCH   64
#define CHANS   3
#define HH      224
#define WW      224
#define GHH     14          // token grid (14x14), one strip = one token row
#define NBINS   9
#define STRIPH  16          // pixel rows per strip (= 16x16 gaussian tile height)
#define TILE_W  (WW + 2)                    // 226 (halo)
#define TILE_H  (STRIPH + 2)                // 18  (halo)
#define TILE_CH (TILE_H * TILE_W)           // 4068 floats per channel
#define TILE_SZ (CHANS * TILE_CH)           // 12204 floats
#define TOKV    108                         // values per token = 27*4
#define HROWS   16                          // histogram padded to 16 token rows
#define HIST_SZ (HROWS * TOKV)              // 1728 floats (rows 14,15 stay zero)
#define NTHREADS 512

typedef __attribute__((ext_vector_type(2))) float v2f;
typedef __attribute__((ext_vector_type(8))) float v8f;

// Direct global -> LDS copy on the CDNA5 async path (ASYNCcnt, bypasses VGPRs).
// lds_off is the byte offset into the (single, offset-0) static LDS block.
__device__ __forceinline__ void async_g2l_b32(unsigned lds_off, const float* gsrc) {
    asm volatile("global_load_async_to_lds_b32 %0, %1, off"
                 :: "v"(lds_off), "v"(gsrc)
                 : "memory");
}
__device__ __forceinline__ void wait_async0() {
    asm volatile("s_wait_asynccnt 0x0" ::: "memory");
}

__global__ __launch_bounds__(NTHREADS)
void hog_fused_kernel(const float* __restrict__ x, float* __restrict__ out) {
    __shared__ float lds[TILE_SZ + HIST_SZ];
    float* tile = lds;
    float* hist = lds + TILE_SZ;

    const int tid = threadIdx.x;
    const int tr  = blockIdx.x;      // token row / strip index: 0..13
    const int b   = blockIdx.y;      // batch

    // ---- zero per-token histograms (all 16 padded rows) ----
    for (int i = tid; i < HIST_SZ; i += NTHREADS) hist[i] = 0.f;

    // ---- async-load strip into LDS with reflect-101 halo (ASYNCcnt path) ----
    const float* xb = x + (size_t)b * CHANS * HH * WW;
    for (int i = tid; i < TILE_SZ; i += NTHREADS) {
        int ch = i / TILE_CH;
        int r  = (i % TILE_CH) / TILE_W;
        int cc = (i % TILE_CH) % TILE_W;
        int gr = tr * STRIPH + r - 1;
        gr = (gr < 0) ? -gr : gr;
        gr = (gr > HH - 1) ? 2 * (HH - 1) - gr : gr;
        int gc = cc - 1;
        gc = (gc < 0) ? -gc : gc;
        gc = (gc > WW - 1) ? 2 * (WW - 1) - gc : gc;
        async_g2l_b32((unsigned)(i * 4), xb + ((size_t)ch * HH + gr) * WW + gc);
    }

    // ---- prefetch next strip into L2 (gfx1250 global_prefetch_b8) ----
    if (tr + 1 < GHH) {
        const int lines = CHANS * (STRIPH * WW / 16);     // one prefetch / 64B
        for (int i = tid; i < lines; i += NTHREADS) {
            int ch = i / (STRIPH * WW / 16);
            int o  = (i % (STRIPH * WW / 16)) * 16;
            __builtin_prefetch(xb + ((size_t)ch * HH + (size_t)(tr + 1) * STRIPH) * WW + o, 0, 3);
        }
    }

    wait_async0();          // all async global->LDS copies complete
    __syncthreads();

    // ---- gaussian normalization: gk(i,j) = w_i * w_j / (sum w)^2, std = 8 ----
    float gs = 0.f;
#pragma unroll
    for (int i = 0; i < 16; ++i) {
        float n = ((float)i - 7.5f) * 0.125f;
        gs += __expf(-0.5f * n * n);
    }
    const float invS2 = 1.f / (gs * gs);
    const float k9opi = 9.0f / 3.14159265358979323846f;

    // ---- per-pixel: sobel -> norm/bin/weight -> LDS-atomic histogram scatter ----
    for (int p = tid; p < CHANS * STRIPH * WW; p += NTHREADS) {
        int ch = p / (STRIPH * WW);
        int r  = (p % (STRIPH * WW)) / WW;     // ly in 0..15
        int cc = p % WW;                       // lx in 0..223
        const float* t0 = tile + ch * TILE_CH + r * TILE_W + cc;  // 3x3 top-left
        float a00 = t0[0],          a01 = t0[1],              a02 = t0[2];
        float a10 = t0[TILE_W],                               a12 = t0[TILE_W + 2];
        float a20 = t0[2 * TILE_W], a21 = t0[2 * TILE_W + 1], a22 = t0[2 * TILE_W + 2];
        float gxv = (a00 + 2.f * a10 + a20) - (a02 + 2.f * a12 + a22);
        float gyv = (a00 + 2.f * a01 + a02) - (a20 + 2.f * a21 + a22);
        float nrm = sqrtf(gxv * gxv + gyv * gyv);
        float ph  = atan2f(gxv, gyv) * k9opi;
        int   bin = (int)floorf(ph) % NBINS;
        if (bin < 0) bin += NBINS;
        int   lx16 = cc & 15;
        float ny = ((float)r    - 7.5f) * 0.125f;
        float nx = ((float)lx16 - 7.5f) * 0.125f;
        float wgt = nrm * __expf(-0.5f * ny * ny) * __expf(-0.5f * nx * nx) * invS2;
        int tok  = cc >> 4;                              // token column 0..13
        int cell = ((r >> 3) << 1) + (lx16 >> 3);        // 2x2 pooled cell
        atomicAdd(&hist[tok * TOKV + (ch * NBINS + bin) * 4 + cell], wgt);
    }
    __syncthreads();

    // ---- token stats via fp32 WMMA on the matrix pipe ----
    // F = hist[16][108] (rows 14,15 zero).  D1 += A * ones -> row sums S1
    //                                       D2 += A * A^T  -> Gram, diag = S2
    // A is 16x4 fp32: lane L holds A[L&15][2*(L>>4) + {0,1}]; the 4x16 B layout
    // is lane-symmetric, so the same VGPR pair serves as both operands of D2.
    // Unconditional ds_load_b64 per chunk (padded rows) -> no exec masking,
    // loads pipeline freely against the 16-cycle wmma cadence.
    float* s1buf = tile;          // reuse tile LDS as scratch (done with it)
    float* s2buf = tile + 256;
    if (tid < 32) {
        const int lane = tid;
        const int m  = lane & 15;               // row (token) 0..15
        const int kb = (lane >> 4) << 1;        // K base: 0 or 2
        v2f ones; ones.x = 1.f; ones.y = 1.f;
        v8f d1 = {0.f, 0.f, 0.f, 0.f, 0.f, 0.f, 0.f, 0.f};
        v8f d2 = {0.f, 0.f, 0.f, 0.f, 0.f, 0.f, 0.f, 0.f};
        const float* hrow = hist + m * TOKV + kb;
#pragma unroll 9
        for (int c = 0; c < 27; ++c) {
            v2f a;
            a.x = hrow[c * 4];
            a.y = hrow[c * 4 + 1];
            d1 = __builtin_amdgcn_wmma_f32_16x16x4_f32(false, a, false, ones,
                                                       (short)0, d1, false, false);
            d2 = __builtin_amdgcn_wmma_f32_16x16x4_f32(false, a, false, a,
                                                       (short)0, d2, false, false);
        }
#pragma unroll
        for (int r = 0; r < 8; ++r) {           // D layout: lanes 0-15 -> M=r, 16-31 -> M=r+8
            int mm = (lane < 16) ? r : (r + 8);
            int nn = lane & 15;
            s1buf[mm * 16 + nn] = d1[r];
            s2buf[mm * 16 + nn] = d2[r];
        }
    }
    __syncthreads();

    // ---- per-token standardize (ddof=1) and mean ----
    if (tid < GHH) {
        float S1   = s1buf[tid * 16];           // any column holds the row sum
        float S2   = s2buf[tid * 16 + tid];     // Gram diagonal
        float mean = S1 * (1.f / 108.f);
        float var  = (S2 - S1 * mean) * (1.f / 107.f);   // unbiased
        float rs   = rsqrtf(var + 1e-6f);
        out[(b * GHH + tr) * GHH + tid] = (S1 - 108.f * mean) * rs * (1.f / 108.f);
    }
}

extern "C" void kernel_launch(void* const* d_in, const int* in_sizes, int n_in,
                              void* d_out, int out_size, void* d_ws, size_t ws_size,
                              hipStream_t stream) {
    (void)in_sizes; (void)n_in; (void)out_size; (void)d_ws; (void)ws_size;
    const float* x = (const float*)d_in[0];   // [64,3,224,224] fp32
    float* out = (float*)d_out;               // [64,196] fp32
    dim3 grid(GHH, BATCH);                    // 14 token-rows x 64 batches
    hog_fused_kernel<<<grid, NTHREADS, 0, stream>>>(x, out);
}